// FourierLayer_90099823935690
// MI455X (gfx1250) — compile-verified
//
#include <hip/hip_runtime.h>
#include <math.h>

// ---------------------------------------------------------------------------
// FNO block for MI455X (gfx1250, wave32, WMMA + async global->LDS).
// B=16, S=2048, H=256, M=32 modes. rfft/irfft replaced by truncated-DFT GEMMs
// through V_WMMA_F32_16X16X4_F32; shared tiles staged in LDS with
// GLOBAL_LOAD_ASYNC_TO_LDS_B128 (ASYNCcnt) to eliminate redundant HBM reads.
// ---------------------------------------------------------------------------

constexpr int B  = 16;
constexpr int S  = 2048;
constexpr int H  = 256;
constexpr int NM = 32;        // retained rfft modes
constexpr int JM = 2 * NM;    // 64 = real rows (0..31) + imag rows (32..63)

typedef __attribute__((ext_vector_type(2))) float v2f;
typedef __attribute__((ext_vector_type(8))) float v8f;

// ---- WMMA f32 16x16x4 wrapper (D = A(16x4) * B(4x16) + C) ------------------
__device__ __forceinline__ v8f wmma4(v2f a, v2f b, v8f c) {
  return __builtin_amdgcn_wmma_f32_16x16x4_f32(
      false, a, false, b, (short)0, c, false, false);
}

// A-style fragment: element (x, k) at P[(x0+x)*ld + k], k contiguous.
__device__ __forceinline__ v2f frag_mk(const float* __restrict__ P, int x0,
                                       int ld, int kb, int lane) {
  int x = x0 + (lane & 15);
  int k = kb + ((lane >> 4) << 1);
  return *(const v2f*)(P + (size_t)x * ld + k);
}

// B-style fragment, k-major stride: element (k, n) at P[k*ld + n0+n].
__device__ __forceinline__ v2f frag_kn(const float* __restrict__ P, int n0,
                                       int ld, int kb, int lane) {
  int n = n0 + (lane & 15);
  int k = kb + ((lane >> 4) << 1);
  v2f b;
  b.x = P[(size_t)k * ld + n];
  b.y = P[(size_t)(k + 1) * ld + n];
  return b;
}

// ---- async global->LDS (CDNA5 ASYNCcnt path) -------------------------------
__device__ __forceinline__ uint32_t lds_addr32(void* p) {
  return (uint32_t)(uintptr_t)(__attribute__((address_space(3))) void*)p;
}
__device__ __forceinline__ void async_g2l_b128(uint32_t lds, const void* g) {
  asm volatile("global_load_async_to_lds_b128 %0, %1, off"
               :: "v"(lds), "v"((unsigned long long)(uintptr_t)g) : "memory");
}
__device__ __forceinline__ void wait_async_le2() {
  asm volatile("s_wait_asynccnt 0x2" ::: "memory");
}
__device__ __forceinline__ void wait_async_0() {
  asm volatile("s_wait_asynccnt 0x0" ::: "memory");
}

// ---------------------------------------------------------------------------
// Kernel 1: DFT bases.
//   DF[j][s], j<32:  cos(2*pi*j*s/S);  j>=32: -sin(2*pi*(j-32)*s/S)
//   CI[s][j], j<32:  c_j*cos(...);     j>=32: -c_k*sin(...)  (c_0=1/S, c_k=2/S)
// ---------------------------------------------------------------------------
__global__ void k_basis(float* __restrict__ DF, float* __restrict__ CI) {
  int idx = blockIdx.x * blockDim.x + threadIdx.x;
  if (idx >= JM * S) return;
  int j = idx / S, s = idx % S;
  int k = j & (NM - 1);
  bool imag = j >= NM;
  int ks = (k * s) & (S - 1);                 // exact mod-2pi reduction
  float ang = (2.0f * (float)M_PI / (float)S) * (float)ks;
  float sv = sinf(ang), cv = cosf(ang);
  DF[(size_t)j * S + s] = imag ? -sv : cv;
  float ck = (k == 0) ? (1.0f / (float)S) : (2.0f / (float)S);
  float coef = imag ? ((k == 0) ? 0.0f : -ck * sv) : ck * cv;
  CI[(size_t)s * JM + j] = coef;
}

// ---------------------------------------------------------------------------
// Kernel 2: forward truncated DFT.  X[b](64x256) = DF(64x2048) @ x[b](2048x256)
// grid (16 n-tiles, 16 batches), block 128 (4 waves; wave w -> row tile 16w).
// The x k-slab (64x16) is shared by all 4 waves -> double-buffered async LDS.
// ---------------------------------------------------------------------------
constexpr int KB = 64;   // k-slab depth

__global__ __launch_bounds__(128) void k_fwd_dft(const float* __restrict__ x,
                                                 const float* __restrict__ DF,
                                                 float* __restrict__ X) {
  int nt = blockIdx.x, b = blockIdx.y;
  int tid = threadIdx.x, wave = tid >> 5, lane = tid & 31;
  int m0 = wave * 16, n0 = nt * 16;
  const float* xb = x + (size_t)b * S * H;

  __shared__ float sB[2][KB * 16];   // [k][n] slab, 4 KB per buffer

  auto stage = [&](int buf, int kb) {   // 256 x 16B quads, 2 per thread
    #pragma unroll
    for (int i = 0; i < 2; ++i) {
      int q = tid + 128 * i;            // 0..255
      int row = q >> 2, quad = (q & 3) * 4;
      const float* src = xb + (size_t)(kb + row) * H + n0 + quad;
      async_g2l_b128(lds_addr32(&sB[buf][row * 16 + quad]), src);
    }
  };

  stage(0, 0);
  v8f c = {};
  for (int kb = 0; kb < S; kb += KB) {
    int cur = (kb / KB) & 1;
    bool has_next = (kb + KB) < S;
    if (has_next) stage(cur ^ 1, kb + KB);   // prefetch next slab
    if (has_next) wait_async_le2();          // oldest slab (2 ops/wave) done
    else          wait_async_0();
    __syncthreads();
    const float* Bs = &sB[cur][0];
    #pragma unroll 4
    for (int kk = 0; kk < KB; kk += 4) {
      v2f a  = frag_mk(DF, m0, S, kb + kk, lane);
      v2f bb = frag_kn(Bs, 0, 16, kk, lane);
      c = wmma4(a, bb, c);
    }
    __syncthreads();   // all reads of `cur` done before it is restaged
  }

  float* Xb = X + (size_t)b * JM * H;
  int col = n0 + (lane & 15);
  int r0  = (lane >> 4) << 3;
  #pragma unroll
  for (int v = 0; v < 8; ++v)
    Xb[(size_t)(m0 + r0 + v) * H + col] = c[v];
}

// ---------------------------------------------------------------------------
// Kernel 3: per-mode complex mixing.  For mode k:
//   YR = XR@WR - XI@WI ;  YI = XR@WI + XI@WR   (16 batches = one M tile)
// grid (16 n-tiles, 32 modes), block 64: wave0 -> YR, wave1 -> YI.
// ---------------------------------------------------------------------------
__global__ __launch_bounds__(64) void k_modemix(const float* __restrict__ X,
                                                const float* __restrict__ wr,
                                                const float* __restrict__ wi,
                                                float* __restrict__ Y) {
  int nt = blockIdx.x, k = blockIdx.y;
  int wave = threadIdx.x >> 5, lane = threadIdx.x & 31;
  int n0 = nt * 16;
  const float* WR = wr + (size_t)k * H * H;
  const float* WI = wi + (size_t)k * H * H;
  const float* B0 = wave ? WI : WR;   // pairs with XR
  const float* B1 = wave ? WR : WI;   // pairs with XI
  const float* XR = X + (size_t)k * H;          // rows stride JM*H (batch-major)
  const float* XI = X + (size_t)(NM + k) * H;
  v8f c0 = {}, c1 = {};
  #pragma unroll 4
  for (int kb = 0; kb < H; kb += 4) {
    v2f aR = frag_mk(XR, 0, JM * H, kb, lane);  // A rows = batches
    v2f aI = frag_mk(XI, 0, JM * H, kb, lane);
    v2f b0 = frag_kn(B0, n0, H, kb, lane);
    v2f b1 = frag_kn(B1, n0, H, kb, lane);
    c0 = wmma4(aR, b0, c0);
    c1 = wmma4(aI, b1, c1);
  }
  int j   = wave ? (NM + k) : k;
  int col = n0 + (lane & 15);
  int r0  = (lane >> 4) << 3;
  #pragma unroll
  for (int v = 0; v < 8; ++v) {
    float val = wave ? (c0[v] + c1[v]) : (c0[v] - c1[v]);
    Y[(size_t)(r0 + v) * JM * H + (size_t)j * H + col] = val;
  }
}

// ---------------------------------------------------------------------------
// Kernel 4 (fused): y = CI@Y[b] + x[b]@conv_w^T + conv_b, then LayerNorm.
// grid (128 s-tiles, 16 batches), block 256 (8 waves; wave w -> n-tiles 2w,2w+1).
// A-tiles (x 16x256 and CI 16x64, shared by all 8 waves) async-staged in LDS
// once; y tile staged in LDS; 16-lane shfl_xor butterfly for mean/var.
// ---------------------------------------------------------------------------
constexpr int LDX = H + 4;    // padded row stride for x / y tiles

__global__ __launch_bounds__(256) void k_out(const float* __restrict__ x,
                                             const float* __restrict__ CI,
                                             const float* __restrict__ Y,
                                             const float* __restrict__ cw,
                                             const float* __restrict__ cb,
                                             const float* __restrict__ gamma,
                                             const float* __restrict__ beta,
                                             float* __restrict__ out) {
  int st = blockIdx.x, b = blockIdx.y;
  int tid = threadIdx.x, wave = tid >> 5, lane = tid & 31;
  int s0 = st * 16;
  const float* xb = x + (size_t)b * S * H;
  const float* Yb = Y + (size_t)b * JM * H;

  __shared__ float smem[16 * LDX + 16 * JM + 16 * LDX];
  float* sX = smem;                 // x tile  [16][LDX]
  float* sC = smem + 16 * LDX;      // CI tile [16][JM]
  float* yt = sC + 16 * JM;         // y tile  [16][LDX]

  {                                  // one-shot async stage of all A data
    int row = tid >> 4, c16 = (tid & 15) * 16;
    #pragma unroll
    for (int q = 0; q < 4; ++q) {    // x tile: 16 KB
      const float* src = xb + (size_t)(s0 + row) * H + c16 + q * 4;
      async_g2l_b128(lds_addr32(&sX[row * LDX + c16 + q * 4]), src);
    }
    int c4 = (tid & 15) * 4;         // CI tile: 4 KB
    async_g2l_b128(lds_addr32(&sC[row * JM + c4]),
                   CI + (size_t)(s0 + row) * JM + c4);
  }
  wait_async_0();
  __syncthreads();

  #pragma unroll
  for (int t = 0; t < 2; ++t) {
    int n0  = (wave * 2 + t) * 16;
    int col = n0 + (lane & 15);
    float bias = cb[col];
    v8f c;
    #pragma unroll
    for (int v = 0; v < 8; ++v) c[v] = bias;
    // inverse DFT: K = 64, A from LDS
    #pragma unroll 4
    for (int kb = 0; kb < JM; kb += 4) {
      v2f a  = frag_mk(sC, 0, JM, kb, lane);
      v2f bb = frag_kn(Yb, n0, H, kb, lane);
      c = wmma4(a, bb, c);
    }
    // 1x1 conv: K = 256, A from LDS; conv_w[o][i] k-contiguous per column
    #pragma unroll 4
    for (int kb = 0; kb < H; kb += 4) {
      v2f a  = frag_mk(sX, 0, LDX, kb, lane);
      v2f bb = frag_mk(cw, n0, H, kb, lane);
      c = wmma4(a, bb, c);
    }
    int r0 = (lane >> 4) << 3;
    #pragma unroll
    for (int v = 0; v < 8; ++v) yt[(r0 + v) * LDX + col] = c[v];
  }
  __syncthreads();

  // LayerNorm: 16 threads per row, 16 values per thread.
  int r  = tid >> 4;
  int c0 = (tid & 15) * 16;
  float vals[16], sum = 0.f, ss = 0.f;
  #pragma unroll
  for (int i = 0; i < 16; ++i) {
    float v = yt[r * LDX + c0 + i];
    vals[i] = v; sum += v; ss += v * v;
  }
  #pragma unroll
  for (int off = 8; off > 0; off >>= 1) {
    sum += __shfl_xor(sum, off, 16);
    ss  += __shfl_xor(ss,  off, 16);
  }
  float mu  = sum * (1.0f / H);
  float var = ss * (1.0f / H) - mu * mu;
  float rs  = rsqrtf(var + 1e-5f);
  float* ob = out + (size_t)b * S * H + (size_t)(s0 + r) * H;
  #pragma unroll
  for (int i = 0; i < 16; ++i) {
    int cc = c0 + i;
    __builtin_nontemporal_store((vals[i] - mu) * rs * gamma[cc] + beta[cc],
                                ob + cc);
  }
}

// ---------------------------------------------------------------------------
extern "C" void kernel_launch(void* const* d_in, const int* in_sizes, int n_in,
                              void* d_out, int out_size, void* d_ws, size_t ws_size,
                              hipStream_t stream) {
  const float* x  = (const float*)d_in[0];
  const float* wr = (const float*)d_in[1];
  const float* wi = (const float*)d_in[2];
  const float* cw = (const float*)d_in[3];
  const float* cb = (const float*)d_in[4];
  const float* gm = (const float*)d_in[5];
  const float* bt = (const float*)d_in[6];
  float* out = (float*)d_out;

  // workspace (floats): DF[64*2048] | CI[2048*64] | X[B*64*H] | Y[B*64*H] = 3 MiB
  float* DF = (float*)d_ws;
  float* CI = DF + (size_t)JM * S;
  float* X  = CI + (size_t)S * JM;
  float* Y  = X + (size_t)B * JM * H;

  k_basis   <<<dim3((JM * S + 255) / 256), 256, 0, stream>>>(DF, CI);
  k_fwd_dft <<<dim3(H / 16, B),  128, 0, stream>>>(x, DF, X);
  k_modemix <<<dim3(H / 16, NM),  64, 0, stream>>>(X, wr, wi, Y);
  k_out     <<<dim3(S / 16, B),  256, 0, stream>>>(x, CI, Y, cw, cb, gm, bt, out);
}